// GNN_38087769981371
// MI455X (gfx1250) — compile-verified
//
#include <hip/hip_runtime.h>
#include <math.h>

#define NNODES 100000
#define NEDGES 1600000
#define D      128
#define NGRAPH 512
#define NCLS   10
#define BN_EPS 1e-5f

#define A_STRIDE 132   // padded row stride (floats) for conflict-free v2f DS reads

typedef __attribute__((ext_vector_type(2))) float v2f;
typedef __attribute__((ext_vector_type(8))) float v8f;

// ---------------- utility kernels ----------------

__global__ void copy_f4(const float4* __restrict__ src, float4* __restrict__ dst, int n4) {
    int i = blockIdx.x * blockDim.x + threadIdx.x;
    if (i < n4) dst[i] = src[i];
}

__global__ void zero_f(float* __restrict__ p, int n) {
    int i = blockIdx.x * blockDim.x + threadIdx.x;
    if (i < n) p[i] = 0.0f;
}

// ---------------- edge scatter-add (L2-resident) ----------------
__global__ void scatter_edges(const float* __restrict__ feat, float* __restrict__ acc,
                              const int* __restrict__ esrc, const int* __restrict__ edst) {
    long long t = (long long)blockIdx.x * blockDim.x + threadIdx.x;
    long long e = t >> 5;
    if (e >= NEDGES) return;
    int f0 = ((int)t & 31) * 4;
    int s = esrc[e];
    int d = edst[e];
    const float4 v = *(const float4*)(feat + (long long)s * D + f0);
    float* dp = acc + (long long)d * D + f0;
    atomicAdd(dp + 0, v.x);
    atomicAdd(dp + 1, v.y);
    atomicAdd(dp + 2, v.z);
    atomicAdd(dp + 3, v.w);
}

// ---------------- fp32 WMMA GEMM: out[M,128] = relu?(A[M,128] @ W[128,128] + b) --------
// Per 256-thread block: A tile (128x128) async-copied to LDS (padded stride 132),
// W swizzled into LDS so each B fragment is a single aligned ds_load_b64.
// Inner loop is branch-free: EXEC is all-ones at every WMMA (ISA requirement).
__global__ __launch_bounds__(256) void gemm_relu(const float* __restrict__ A,
                                                 const float* __restrict__ W,
                                                 const float* __restrict__ bias,
                                                 float* __restrict__ out,
                                                 int M, int doRelu) {
    // Swizzled W: Wl[(k>>2)*512 + c*4 + (k&3)] = W[k][c]  (64 KB, conflict-free reads)
    __shared__ float Wl[D * D];
    // Padded A tile: Al[r*132 + c] = A[rowBase0 + r][c]  (~66 KB, conflict-free reads)
    __shared__ float Al[128 * A_STRIDE];

    const int rowBase0 = blockIdx.x * 128;

    // ---- stage W (synchronous float4 loads, swizzled scalar LDS stores) ----
    for (int i = threadIdx.x; i < (D * D) / 4; i += 256) {
        const int kk = i >> 5;          // W row (k index)
        const int c0 = (i & 31) * 4;    // W col group
        const float4 w4 = ((const float4*)W)[i];
        float* dst = &Wl[(kk >> 2) * 512 + c0 * 4 + (kk & 3)];
        dst[0]  = w4.x;
        dst[4]  = w4.y;
        dst[8]  = w4.z;
        dst[12] = w4.w;
    }

    // ---- stage A tile via async global->LDS (ASYNCcnt path), 16B per lane ----
    for (int i = threadIdx.x; i < 128 * (D / 4); i += 256) {
        const int r  = i >> 5;          // tile row
        const int c0 = (i & 31) * 4;    // col group
        if (rowBase0 + r < M) {
            unsigned lds_off = (unsigned)(size_t)(&Al[r * A_STRIDE + c0]);
            unsigned long long gaddr =
                (unsigned long long)(const void*)(A + (long long)(rowBase0 + r) * D + c0);
            asm volatile("global_load_async_to_lds_b128 %0, %1, off"
                         :: "v"(lds_off), "v"(gaddr) : "memory");
        }
    }
    asm volatile("s_wait_asynccnt 0" ::: "memory");
    __syncthreads();

    const int lane = threadIdx.x & 31;
    const int wave = threadIdx.x >> 5;
    const int lc   = lane & 15;        // col-in-tile / row-in-stripe
    const int half = lane >> 4;        // 0: K={0,1}, 1: K={2,3}
    const int koff = half << 1;
    const int rowBase = rowBase0 + wave * 16;

    v8f acc[8];
#pragma unroll
    for (int nt = 0; nt < 8; ++nt) acc[nt] = (v8f)0.0f;

    const float* aRow = &Al[(wave * 16 + lc) * A_STRIDE + koff];
    const float* wRow = &Wl[lc * 4 + koff];

    for (int k = 0; k < D; k += 4) {
        // A fragment: one conflict-free ds_load_b64
        const v2f a = *(const v2f*)(aRow + k);
        const float* wk = wRow + (k >> 2) * 512;
#pragma unroll
        for (int nt = 0; nt < 8; ++nt) {
            // B fragment: one conflict-free ds_load_b64 (k-interleaved swizzle)
            const v2f b = *(const v2f*)(wk + nt * 64);
            acc[nt] = __builtin_amdgcn_wmma_f32_16x16x4_f32(
                false, a, false, b, (short)0, acc[nt], false, false);
        }
    }

    // C/D layout: VGPR v -> row = rowBase + v + 8*half, col = nt*16 + lc
#pragma unroll
    for (int nt = 0; nt < 8; ++nt) {
        const int col = nt * 16 + lc;
        const float bc = bias[col];
#pragma unroll
        for (int v = 0; v < 8; ++v) {
            const int r = rowBase + v + (half << 3);
            if (r < M) {
                float val = acc[nt][v] + bc;
                if (doRelu) val = fmaxf(val, 0.0f);
                out[(long long)r * D + col] = val;
            }
        }
    }
}

// ---------------- graph pooling scatter ----------------
__global__ void pool_scatter(const float* __restrict__ h, const int* __restrict__ gidx,
                             float* __restrict__ pooled) {
    long long t = (long long)blockIdx.x * blockDim.x + threadIdx.x;
    if (t >= (long long)NNODES * D) return;
    int n = (int)(t / D);
    int f = (int)(t % D);
    atomicAdd(&pooled[(long long)gidx[n] * D + f], h[t]);
}

// ---------------- batchnorm over graph dim (training-mode batch stats) ----------------
__global__ void bn_kernel(const float* __restrict__ pooled, const float* __restrict__ gamma,
                          const float* __restrict__ beta, float* __restrict__ outv) {
    int c = threadIdx.x;  // 128 threads, one per feature
    float s = 0.0f, ss = 0.0f;
    for (int g = 0; g < NGRAPH; ++g) {
        float v = pooled[g * D + c];
        s += v;
        ss += v * v;
    }
    const float mean = s * (1.0f / NGRAPH);
    const float var  = ss * (1.0f / NGRAPH) - mean * mean;
    const float scl  = rsqrtf(var + BN_EPS) * gamma[c];
    const float bt   = beta[c];
    for (int g = 0; g < NGRAPH; ++g) {
        outv[g * D + c] = (pooled[g * D + c] - mean) * scl + bt;
    }
}

// ---------------- head: logits = zg @ W4 + b4 ; log_softmax ----------------
__global__ void head_kernel(const float* __restrict__ zg, const float* __restrict__ W4,
                            const float* __restrict__ b4, float* __restrict__ out) {
    int g = blockIdx.x * blockDim.x + threadIdx.x;
    if (g >= NGRAPH) return;
    float logits[NCLS];
#pragma unroll
    for (int c = 0; c < NCLS; ++c) logits[c] = b4[c];
    const float* z = zg + (long long)g * D;
    for (int k = 0; k < D; ++k) {
        float zk = z[k];
#pragma unroll
        for (int c = 0; c < NCLS; ++c) logits[c] += zk * W4[k * NCLS + c];
    }
    float mx = logits[0];
#pragma unroll
    for (int c = 1; c < NCLS; ++c) mx = fmaxf(mx, logits[c]);
    float se = 0.0f;
#pragma unroll
    for (int c = 0; c < NCLS; ++c) se += expf(logits[c] - mx);
    const float lse = mx + logf(se);
#pragma unroll
    for (int c = 0; c < NCLS; ++c) out[g * NCLS + c] = logits[c] - lse;
}

// ---------------- launch ----------------
extern "C" void kernel_launch(void* const* d_in, const int* in_sizes, int n_in,
                              void* d_out, int out_size, void* d_ws, size_t ws_size,
                              hipStream_t stream) {
    const float* x     = (const float*)d_in[0];
    const int*   eidx  = (const int*)d_in[1];
    const int*   esrc  = eidx;
    const int*   edst  = eidx + NEDGES;
    const int*   gidx  = (const int*)d_in[2];
    const float* W1    = (const float*)d_in[3];
    const float* b1    = (const float*)d_in[4];
    const float* W2    = (const float*)d_in[5];
    const float* b2    = (const float*)d_in[6];
    const float* W3    = (const float*)d_in[7];
    const float* b3    = (const float*)d_in[8];
    const float* W4    = (const float*)d_in[9];
    const float* b4    = (const float*)d_in[10];
    const float* gamma = (const float*)d_in[11];
    const float* beta  = (const float*)d_in[12];
    float* outp = (float*)d_out;

    float* bufA   = (float*)d_ws;                       // [N, D]
    float* bufB   = bufA + (size_t)NNODES * D;          // [N, D]
    float* pooled = bufB + (size_t)NNODES * D;          // [G, D]
    float* bn     = pooled + (size_t)NGRAPH * D;        // [G, D]
    float* zg     = bn + (size_t)NGRAPH * D;            // [G, D]

    const int NH4 = (NNODES * D) / 4;
    const long long ethreads = (long long)NEDGES * 32;
    const int eblocks = (int)((ethreads + 255) / 256);
    const long long pthreads = (long long)NNODES * D;
    const int pblocks = (int)((pthreads + 255) / 256);
    const int gemmNodeBlocks = (NNODES + 127) / 128;

    // layer 1: agg = x + scatter(x); z1 = relu(agg @ W1 + b1)
    copy_f4<<<(NH4 + 255) / 256, 256, 0, stream>>>((const float4*)x, (float4*)bufA, NH4);
    scatter_edges<<<eblocks, 256, 0, stream>>>(x, bufA, esrc, edst);
    gemm_relu<<<gemmNodeBlocks, 256, 0, stream>>>(bufA, W1, b1, bufB, NNODES, 1);

    // layer 2: agg2 = z1 + scatter(z1); h = agg2 @ W2 + b2
    copy_f4<<<(NH4 + 255) / 256, 256, 0, stream>>>((const float4*)bufB, (float4*)bufA, NH4);
    scatter_edges<<<eblocks, 256, 0, stream>>>(bufB, bufA, esrc, edst);
    gemm_relu<<<gemmNodeBlocks, 256, 0, stream>>>(bufA, W2, b2, bufB, NNODES, 0);

    // pooled readout + batchnorm
    zero_f<<<(NGRAPH * D + 255) / 256, 256, 0, stream>>>(pooled, NGRAPH * D);
    pool_scatter<<<pblocks, 256, 0, stream>>>(bufB, gidx, pooled);
    bn_kernel<<<1, D, 0, stream>>>(pooled, gamma, beta, bn);

    // graph MLP head
    gemm_relu<<<(NGRAPH + 127) / 128, 256, 0, stream>>>(bn, W3, b3, zg, NGRAPH, 1);
    head_kernel<<<(NGRAPH + 255) / 256, 256, 0, stream>>>(zg, W4, b4, outp);
}